// AlignGNN_v2_2783138808234
// MI455X (gfx1250) — compile-verified
//
#include <hip/hip_runtime.h>

// Problem constants (from the reference):
#define BB   4
#define NN   10000
#define EE   160000
#define MM   4
#define HH   64
#define OUTF 64
#define KK   128                 // 2H
#define ROWS (BB * NN * MM)      // 160000 rows of the fused GEMM
#define CH   (MM * OUTF)         // 256 floats per (b, node)

typedef __attribute__((ext_vector_type(2))) float v2f;
typedef __attribute__((ext_vector_type(8))) float v8f;

// -------------------------------------------------------------------------
// Hardware f32 atomic add, fire-and-forget, device scope (out fits in L2,
// so these RMW in GL2 at L2 bandwidth). Inline asm avoids any CAS expansion.
// -------------------------------------------------------------------------
__device__ __forceinline__ void gatomic_add_f32(float* p, float v) {
    asm volatile("global_atomic_add_f32 %0, %1, off scope:SCOPE_DEV"
                 :: "v"(p), "v"(v) : "memory");
}

// -------------------------------------------------------------------------
// Kernel 0: zero the output accumulator (harness poisons it with 0xAA).
// -------------------------------------------------------------------------
__global__ void zero_f32_kernel(float* __restrict__ p, int n4) {
    float4 z = make_float4(0.f, 0.f, 0.f, 0.f);
    for (int i = blockIdx.x * blockDim.x + threadIdx.x; i < n4;
         i += gridDim.x * blockDim.x)
        ((float4*)p)[i] = z;
}

// -------------------------------------------------------------------------
// Kernel 1: nh_fts[r,o] = sum_k A[r,k] * W_nh[k,o] + b_nh[o]
//   A[r,k] = (k < 64) ? node[r*64+k] : hidden[r*64+(k-64)]   (virtual concat)
// One wave -> 16x64 output strip via V_WMMA_F32_16X16X4_F32, 32 k-steps.
// W_nh (128x64 = 32KB) staged transposed in LDS, pitch 132 floats so that
// the 16 lanes of a b-frag read hit banks n*132*4/4 % 64 = 4n (conflict-free).
// -------------------------------------------------------------------------
#define WPITCH 132
__global__ __launch_bounds__(256)
void gemm_wmma_kernel(const float* __restrict__ node,
                      const float* __restrict__ hidden,
                      const float* __restrict__ W,      // [128,64] row-major
                      const float* __restrict__ bias,   // [64]
                      float*       __restrict__ D) {    // [ROWS,64]
    __shared__ float lw[OUTF * WPITCH];                 // lw[o*WPITCH + k] = W[k,o]

    const int tid = threadIdx.x;
    for (int idx = tid; idx < KK * OUTF; idx += 256) {  // coalesced read, once
        int k = idx >> 6, o = idx & 63;
        lw[o * WPITCH + k] = W[idx];
    }
    __syncthreads();

    const int wave = tid >> 5;
    const int lane = tid & 31;
    const int half = lane >> 4;      // 0: K pair {k,k+1}, 1: K pair {k+2,k+3}
    const int n16  = lane & 15;      // M for A-frag, N for B/C/D frags

    const int r0 = blockIdx.x * 128 + wave * 16;        // 1250*128 == ROWS
    const float* arow_n = node   + (long long)(r0 + n16) * HH;
    const float* arow_h = hidden + (long long)(r0 + n16) * HH;

    v8f c0 = {}, c1 = {}, c2 = {}, c3 = {};

    for (int ks = 0; ks < 32; ++ks) {
        const int kk = ks * 4 + half * 2;               // even; pair {kk,kk+1}
        // A-frag: float2 from the virtually-concatenated row (kk pair never
        // straddles the node/hidden boundary since 64 % 4 == 0).
        const float* ap = (kk < HH) ? (arow_n + kk) : (arow_h + (kk - HH));
        v2f a = *(const v2f*)ap;
        // B-frags for the 4 column tiles (8-byte aligned LDS reads).
        v2f b0 = *(const v2f*)&lw[( 0 + n16) * WPITCH + kk];
        v2f b1 = *(const v2f*)&lw[(16 + n16) * WPITCH + kk];
        v2f b2 = *(const v2f*)&lw[(32 + n16) * WPITCH + kk];
        v2f b3 = *(const v2f*)&lw[(48 + n16) * WPITCH + kk];

        c0 = __builtin_amdgcn_wmma_f32_16x16x4_f32(false, a, false, b0, (short)0, c0, false, false);
        c1 = __builtin_amdgcn_wmma_f32_16x16x4_f32(false, a, false, b1, (short)0, c1, false, false);
        c2 = __builtin_amdgcn_wmma_f32_16x16x4_f32(false, a, false, b2, (short)0, c2, false, false);
        c3 = __builtin_amdgcn_wmma_f32_16x16x4_f32(false, a, false, b3, (short)0, c3, false, false);
    }

    // Bias + store. D/C layout: row = vgpr + 8*half, col = tile*16 + n16.
    const float bs0 = bias[ 0 + n16];
    const float bs1 = bias[16 + n16];
    const float bs2 = bias[32 + n16];
    const float bs3 = bias[48 + n16];
    #pragma unroll
    for (int i = 0; i < 8; ++i) {
        float* dr = D + (long long)(r0 + i + 8 * half) * OUTF + n16;
        dr[ 0] = c0[i] + bs0;
        dr[16] = c1[i] + bs1;
        dr[32] = c2[i] + bs2;
        dr[48] = c3[i] + bs3;
    }
}

// -------------------------------------------------------------------------
// Kernel 2: fused edge gate + gather + atomic scatter. One wave per edge.
//   coeff = dot(edge_fts[b,e,:], W_e) + b_e          (wave reduction)
//   out[b,tgt,:,:] += coeff * nh_fts[b,src,:,:]       (256 floats, 8/lane)
// nh_fts (41MB) and out (41MB) are L2-resident on MI455X (192MB L2), so the
// random gather and the f32 atomics run at L2 bandwidth; edge_fts streams.
// -------------------------------------------------------------------------
__global__ __launch_bounds__(256)
void edge_scatter_kernel(const float*     __restrict__ edge_fts, // [B*E,64]
                         const float*     __restrict__ We,       // [64]
                         const float*     __restrict__ be,       // [1]
                         const long long* __restrict__ eidx,     // [B*E,2] int64
                         const float*     __restrict__ nh,       // [B,N,256]
                         float*           __restrict__ out) {    // [B,N,256]
    const long long gtid = (long long)blockIdx.x * blockDim.x + threadIdx.x;
    const long long wv   = gtid >> 5;                  // edge id (flat b*E+e)
    const int       lane = threadIdx.x & 31;
    if (wv >= (long long)BB * EE) return;

    const int b = (int)(wv / EE);

    // Gate: 64-element dot product, 2 floats per lane, xor-shuffle reduce.
    const float2 ef = ((const float2*)(edge_fts + wv * 64))[lane];
    const float2 w  = ((const float2*)We)[lane];
    float part = ef.x * w.x + ef.y * w.y;
    #pragma unroll
    for (int off = 16; off > 0; off >>= 1)
        part += __shfl_xor(part, off, 32);
    const float coeff = part + be[0];

    const long long s = eidx[wv * 2 + 0];
    const long long t = eidx[wv * 2 + 1];

    const float4* sp = (const float4*)(nh + ((long long)b * NN + s) * CH + lane * 8);
    float4 u = sp[0];
    float4 v = sp[1];
    float* dp = out + ((long long)b * NN + t) * CH + lane * 8;

    gatomic_add_f32(dp + 0, u.x * coeff);
    gatomic_add_f32(dp + 1, u.y * coeff);
    gatomic_add_f32(dp + 2, u.z * coeff);
    gatomic_add_f32(dp + 3, u.w * coeff);
    gatomic_add_f32(dp + 4, v.x * coeff);
    gatomic_add_f32(dp + 5, v.y * coeff);
    gatomic_add_f32(dp + 6, v.z * coeff);
    gatomic_add_f32(dp + 7, v.w * coeff);
}

// -------------------------------------------------------------------------
extern "C" void kernel_launch(void* const* d_in, const int* in_sizes, int n_in,
                              void* d_out, int out_size, void* d_ws, size_t ws_size,
                              hipStream_t stream) {
    // setup_inputs order:
    const float*     node_fts = (const float*)d_in[0];   // [B,N,M,H]
    const float*     hidden   = (const float*)d_in[1];   // [B,N,M,H]
    const float*     edge_fts = (const float*)d_in[2];   // [B,E,F]
    const float*     W_nh     = (const float*)d_in[3];   // [2H,OUT]
    const float*     b_nh     = (const float*)d_in[4];   // [OUT]
    const float*     W_e      = (const float*)d_in[5];   // [F,1]
    const float*     b_e      = (const float*)d_in[6];   // [1]
    const long long* eidx     = (const long long*)d_in[7]; // [B,E,2] int64

    float* out = (float*)d_out;                // [B,N,M,OUT] = 10,240,000 f32
    float* nh  = (float*)d_ws;                 // nh_fts scratch: 40.96 MB

    // 0) zero the scatter accumulator
    zero_f32_kernel<<<2560, 256, 0, stream>>>(out, out_size / 4);

    // 1) WMMA GEMM: 160000 rows / (8 waves * 16 rows) = 1250 blocks
    gemm_wmma_kernel<<<ROWS / 128, 256, 0, stream>>>(node_fts, hidden, W_nh, b_nh, nh);

    // 2) fused gate + gather + atomic scatter: one wave per edge
    const long long nedges = (long long)BB * EE;       // 640000 waves
    const int blocks = (int)((nedges * 32 + 255) / 256);
    edge_scatter_kernel<<<blocks, 256, 0, stream>>>(edge_fts, W_e, b_e, eidx, nh, out);
}